// Reconstruction_53025666237081
// MI455X (gfx1250) — compile-verified
//
#include <hip/hip_runtime.h>
#include <hip/hip_bf16.h>
#include <math.h>

typedef __attribute__((ext_vector_type(2))) float v2f;
typedef __attribute__((ext_vector_type(8))) float v8f;

#define NVOX   110592        // 48*48*48
#define TILE   128           // voxels per workgroup (8 waves * 16)
#define KTOT   92            // 90 equi rows + 2 inva rows
#define KSTEPS 23            // 92 / 4
#define MROWS  192           // S*Vs
#define SIN_STRIDE 136       // padded row stride (dwords) for bank-conflict-free B reads
#define OUT1_ELEMS ((size_t)MROWS * NVOX)

// ---------------------------------------------------------------------------
// Setup: build fused weight matrix W[192][92] and fe scale table fev[270]
//   W[s*64+v][n*45+c] = Y[s,v,c] * pfe[n,s,l(c)] * sqrt(4pi/(4l+1))
//   W[s*64+v][90+n]   = (sum_c Y[s,v,c]) * pfi[n,s,0] * sqrt(4pi)
//   fev[(n*3+s)*45+c] = pfe[n,s,l(c)] * sqrt(4pi/(4l+1))
// ---------------------------------------------------------------------------
__global__ __launch_bounds__(256) void Reconstruction_setup_kernel(
    const float* __restrict__ pfe,   // (2,3,5)
    const float* __restrict__ pfi,   // (2,3,1)
    const float* __restrict__ Y,     // (3,64,45)
    float* __restrict__ W,           // [192][92]
    float* __restrict__ fev)         // [270]
{
    const int tid = threadIdx.x;
    const float PI4 = 4.0f * 3.14159265358979323846f;

    if (tid < MROWS) {
        const int s = tid / 64;
        float sumY = 0.0f;
        for (int c = 0; c < 45; ++c) {
            const float y = Y[(size_t)tid * 45 + c];
            sumY += y;
            const int l = (c < 1) ? 0 : (c < 6) ? 1 : (c < 15) ? 2 : (c < 28) ? 3 : 4;
            const float sc = sqrtf(PI4 / (float)(4 * l + 1));
            for (int n = 0; n < 2; ++n)
                W[(size_t)tid * KTOT + n * 45 + c] = y * pfe[(n * 3 + s) * 5 + l] * sc;
        }
        const float sc0 = sqrtf(PI4);
        for (int n = 0; n < 2; ++n)
            W[(size_t)tid * KTOT + 90 + n] = sumY * pfi[n * 3 + s] * sc0;
    }

    for (int idx = tid; idx < 270; idx += blockDim.x) {
        const int n = idx / 135;
        const int rem = idx % 135;
        const int s = rem / 45;
        const int c = rem % 45;
        const int l = (c < 1) ? 0 : (c < 6) ? 1 : (c < 15) ? 2 : (c < 28) ? 3 : 4;
        const float sc = sqrtf(PI4 / (float)(4 * l + 1));
        fev[idx] = pfe[(n * 3 + s) * 5 + l] * sc;
    }
}

// ---------------------------------------------------------------------------
// Fused kernel: one pass over voxels.
//  Phase 1: stage input column tile (92 x 128) into LDS; while each equi value
//           is live in a register, emit the 3 scaled out2 stores (NT).
//  Phase 2: per-wave 16-voxel column GEMM via V_WMMA_F32_16X16X4_F32.
//           B-frags (23 x v2f) live in registers for the whole wave; A-frags
//           are double-buffered across M-tile pairs so the 23-WMMA accumulate
//           chain runs without per-pair s_wait_dscnt stalls.
// ---------------------------------------------------------------------------
__global__ __launch_bounds__(256) void Reconstruction_main_kernel(
    const float* __restrict__ xe,    // (2,45,NVOX) equi
    const float* __restrict__ xi,    // (2,NVOX)    inva
    const float* __restrict__ W,     // [192][92]
    const float* __restrict__ fev,   // [270]
    float* __restrict__ out1,        // (192, NVOX)
    float* __restrict__ out2)        // (270, NVOX)
{
    extern __shared__ float smem[];
    float* sW  = smem;                       // 192*92   = 17664 floats
    float* sIn = smem + MROWS * KTOT;        // 92*136   = 12512 floats

    const int tid = threadIdx.x;
    const int tileBase = blockIdx.x * TILE;

    // ---- stage weights (hits L2 after first block) ----
    for (int i = tid; i < MROWS * KTOT; i += 256)
        sW[i] = W[i];

    // ---- stage inputs + produce out2 ----
    #pragma unroll 2
    for (int it = 0; it < (KTOT * TILE) / 256; ++it) {
        const int idx = it * 256 + tid;
        const int row = idx >> 7;            // 0..91
        const int col = idx & 127;
        const int p = tileBase + col;
        float val;
        if (row < 90) {
            val = __builtin_nontemporal_load(xe + (size_t)row * NVOX + p);
            const int n = row / 45;
            const int c = row - n * 45;
            #pragma unroll
            for (int s = 0; s < 3; ++s) {
                const int f = (n * 3 + s) * 45 + c;
                __builtin_nontemporal_store(fev[f] * val, out2 + (size_t)f * NVOX + p);
            }
        } else {
            val = __builtin_nontemporal_load(xi + (size_t)(row - 90) * NVOX + p);
        }
        sIn[row * SIN_STRIDE + col] = val;
    }
    __syncthreads();

    // ---- GEMM phase: wave w owns voxel column [cb, cb+16) ----
    const int wave = tid >> 5;
    const int lane = tid & 31;
    const int half = lane >> 4;              // K half-select (A & B)
    const int li   = lane & 15;              // M (A) / N (B) index
    const int cb   = wave * 16;
    const int pc   = tileBase + cb + li;

    // B fragments: loaded once, held in 46 VGPRs for all 12 M-tiles.
    const float* sB = sIn + (size_t)(half * 2) * SIN_STRIDE + cb + li;
    v2f bfr[KSTEPS];
    #pragma unroll
    for (int ks = 0; ks < KSTEPS; ++ks) {
        v2f b;
        b.x = sB[(ks * 4    ) * SIN_STRIDE];
        b.y = sB[(ks * 4 + 1) * SIN_STRIDE];
        bfr[ks] = b;
    }

    // A fragments: ping-pong double buffer across M-tile pairs.
    const float* sA = sW + (size_t)li * KTOT + half * 2;   // even -> 8B aligned
    v2f aE[KSTEPS], aO[KSTEPS];
    #pragma unroll
    for (int ks = 0; ks < KSTEPS; ++ks)
        aE[ks] = *(const v2f*)(sA + ks * 4);

    for (int mt = 0; mt < 12; mt += 2) {
        // issue loads for tile mt+1 while tile mt's WMMA chain executes
        const float* sA1 = sA + (size_t)(mt + 1) * 16 * KTOT;
        #pragma unroll
        for (int ks = 0; ks < KSTEPS; ++ks)
            aO[ks] = *(const v2f*)(sA1 + ks * 4);

        v8f acc = {0.f, 0.f, 0.f, 0.f, 0.f, 0.f, 0.f, 0.f};
        #pragma unroll
        for (int ks = 0; ks < KSTEPS; ++ks)
            acc = __builtin_amdgcn_wmma_f32_16x16x4_f32(
                false, aE[ks], false, bfr[ks], (short)0, acc, false, false);
        {
            const int rowBase = mt * 16 + half * 8;  // D: VGPR r -> M=r / r+8
            #pragma unroll
            for (int r = 0; r < 8; ++r)
                __builtin_nontemporal_store(acc[r],
                    out1 + (size_t)(rowBase + r) * NVOX + pc);
        }

        // issue loads for tile mt+2 while tile mt+1's WMMA chain executes
        if (mt + 2 < 12) {
            const float* sA2 = sA + (size_t)(mt + 2) * 16 * KTOT;
            #pragma unroll
            for (int ks = 0; ks < KSTEPS; ++ks)
                aE[ks] = *(const v2f*)(sA2 + ks * 4);
        }

        v8f acc2 = {0.f, 0.f, 0.f, 0.f, 0.f, 0.f, 0.f, 0.f};
        #pragma unroll
        for (int ks = 0; ks < KSTEPS; ++ks)
            acc2 = __builtin_amdgcn_wmma_f32_16x16x4_f32(
                false, aO[ks], false, bfr[ks], (short)0, acc2, false, false);
        {
            const int rowBase = (mt + 1) * 16 + half * 8;
            #pragma unroll
            for (int r = 0; r < 8; ++r)
                __builtin_nontemporal_store(acc2[r],
                    out1 + (size_t)(rowBase + r) * NVOX + pc);
        }
    }
}

extern "C" void kernel_launch(void* const* d_in, const int* in_sizes, int n_in,
                              void* d_out, int out_size, void* d_ws, size_t ws_size,
                              hipStream_t stream) {
    const float* xe  = (const float*)d_in[0];   // x_equi_shc  (1,2,45,48,48,48)
    const float* xi  = (const float*)d_in[1];   // x_inva_shc  (1,2,1,48,48,48)
    const float* pfe = (const float*)d_in[2];   // polar_filter_equi (2,3,5)
    const float* pfi = (const float*)d_in[3];   // polar_filter_inva (2,3,1)
    const float* Y   = (const float*)d_in[4];   // (3,64,45)

    float* out1 = (float*)d_out;                // x_reconstructed (1,192,48^3)
    float* out2 = out1 + OUT1_ELEMS;            // x_equi_conv_shc (1,2,3,45,48^3)

    float* W   = (float*)d_ws;                  // 192*92 floats
    float* fev = W + MROWS * KTOT;              // 270 floats

    Reconstruction_setup_kernel<<<1, 256, 0, stream>>>(pfe, pfi, Y, W, fev);

    const size_t smem_bytes = (size_t)(MROWS * KTOT + KTOT * SIN_STRIDE) * sizeof(float);
    Reconstruction_main_kernel<<<NVOX / TILE, 256, smem_bytes, stream>>>(
        xe, xi, W, fev, out1, out2);
}